// GATCustom_72834055405954
// MI455X (gfx1250) — compile-verified
//
#include <hip/hip_runtime.h>

#define N_BATCH 32
#define N_NODES 256
#define N_IN    128
#define N_OUT   64
#define ALPHA   0.3f
#define S_LD    192   // S = [ s_i(64) | s_j(64) | h(64) ] per row

typedef float v2f __attribute__((ext_vector_type(2)));
typedef float v8f __attribute__((ext_vector_type(8)));

// ---------------------------------------------------------------------------
// K1: S[8192,192] = X[8192,128] @ [ Wl[:128] | Wl[128:] | W ]   (fp32 WMMA)
// One wave per 64x16 output block (4 row-tiles), so the strided B fragment
// is loaded once per k-step and reused by 4 WMMAs.
// ---------------------------------------------------------------------------
__global__ __launch_bounds__(32)
void gat_proj_wmma(const float* __restrict__ X,
                   const float* __restrict__ W,
                   const float* __restrict__ Wl,
                   float* __restrict__ S)
{
    const int lane = threadIdx.x & 31;
    const int half = lane >> 4;      // 0: lanes 0-15, 1: lanes 16-31
    const int l16  = lane & 15;
    const int rowBlk  = blockIdx.x;  // 0..127  (64 rows each)
    const int colTile = blockIdx.y;  // 0..11
    const int g  = colTile >> 2;     // 0: s_i weights, 1: s_j weights, 2: W
    const int nb = (colTile & 3) * 16;

    const float* Bp; int rowOff;
    if      (g == 0) { Bp = Wl; rowOff = 0;   }
    else if (g == 1) { Bp = Wl; rowOff = 128; }
    else             { Bp = W;  rowOff = 0;   }

    // A fragment: M = l16, K components = kk + {0,1}  (ISA 7.12.2)
    const float* arow0 = X + (long)(rowBlk * 64 + l16) * N_IN;

    v8f c0 = {0.f,0.f,0.f,0.f,0.f,0.f,0.f,0.f};
    v8f c1 = c0, c2 = c0, c3 = c0;

    #pragma unroll 2
    for (int k0 = 0; k0 < N_IN; k0 += 4) {
        const int kk = k0 + 2 * half;
        // B fragment: N = l16, K components = kk + {0,1} — shared by 4 WMMAs
        v2f b;
        b.x = Bp[(rowOff + kk    ) * N_OUT + nb + l16];
        b.y = Bp[(rowOff + kk + 1) * N_OUT + nb + l16];

        v2f a0, a1, a2, a3;
        a0.x = arow0[kk];              a0.y = arow0[kk + 1];
        a1.x = arow0[16 * N_IN + kk];  a1.y = arow0[16 * N_IN + kk + 1];
        a2.x = arow0[32 * N_IN + kk];  a2.y = arow0[32 * N_IN + kk + 1];
        a3.x = arow0[48 * N_IN + kk];  a3.y = arow0[48 * N_IN + kk + 1];

        c0 = __builtin_amdgcn_wmma_f32_16x16x4_f32(false, a0, false, b, (short)0, c0, false, false);
        c1 = __builtin_amdgcn_wmma_f32_16x16x4_f32(false, a1, false, b, (short)0, c1, false, false);
        c2 = __builtin_amdgcn_wmma_f32_16x16x4_f32(false, a2, false, b, (short)0, c2, false, false);
        c3 = __builtin_amdgcn_wmma_f32_16x16x4_f32(false, a3, false, b, (short)0, c3, false, false);
    }

    // C/D layout: component v -> M = v + 8*half, N = l16
    float* dst = S + (long)(rowBlk * 64 + 8 * half) * S_LD + colTile * 16 + l16;
    #pragma unroll
    for (int v = 0; v < 8; ++v) dst[(v     ) * S_LD] = c0[v];
    #pragma unroll
    for (int v = 0; v < 8; ++v) dst[(v + 16) * S_LD] = c1[v];
    #pragma unroll
    for (int v = 0; v < 8; ++v) dst[(v + 32) * S_LD] = c2[v];
    #pragma unroll
    for (int v = 0; v < 8; ++v) dst[(v + 48) * S_LD] = c3[v];
}

// ---------------------------------------------------------------------------
// K2: att[b,i,j] = sum_d lrelu(s_i[i,d]+s_j[j,d]) * a[d]; row softmax -> P.
// One block (8 waves) per graph. s_j staged transposed in LDS: sjT[d][j]
// so lanes-over-j reads are bank-conflict-free (64 banks, 32 lanes).
// ---------------------------------------------------------------------------
__global__ __launch_bounds__(256)
void gat_att_softmax(const float* __restrict__ S,
                     const float* __restrict__ avec,
                     float* __restrict__ P)
{
    extern __shared__ float sjT[];   // [64][256] = 64 KB
    const int b = blockIdx.x;
    const int t = threadIdx.x;
    const float* Sb = S + (long)b * N_NODES * S_LD;

    for (int idx = t; idx < N_OUT * N_NODES; idx += 256) {
        const int d = idx >> 8;          // 0..63
        const int j = idx & 255;         // 0..255
        sjT[idx] = Sb[(long)j * S_LD + 64 + d];
    }
    __syncthreads();

    const int lane = t & 31;
    const int wave = t >> 5;

    for (int i = wave; i < N_NODES; i += 8) {
        const float* si = Sb + (long)i * S_LD;   // s_i row, cols 0..63
        float acc[8];
        #pragma unroll
        for (int jj = 0; jj < 8; ++jj) acc[jj] = 0.f;

        for (int d = 0; d < N_OUT; ++d) {
            const float sid = si[d];             // wave-uniform
            const float ad  = avec[d];           // wave-uniform
            const float* row = sjT + d * N_NODES + lane;
            #pragma unroll
            for (int jj = 0; jj < 8; ++jj) {
                float z = sid + row[jj * 32];
                z = fmaxf(z, ALPHA * z);         // leaky_relu, 0<alpha<1
                acc[jj] = fmaf(z, ad, acc[jj]);
            }
        }

        // softmax over 256 j values (8 per lane x 32 lanes, wave32 reduce)
        float m = acc[0];
        #pragma unroll
        for (int jj = 1; jj < 8; ++jj) m = fmaxf(m, acc[jj]);
        for (int o = 16; o > 0; o >>= 1) m = fmaxf(m, __shfl_xor(m, o, 32));

        float p[8], s = 0.f;
        #pragma unroll
        for (int jj = 0; jj < 8; ++jj) { p[jj] = __expf(acc[jj] - m); s += p[jj]; }
        for (int o = 16; o > 0; o >>= 1) s += __shfl_xor(s, o, 32);
        const float inv = 1.0f / s;

        float* Pr = P + ((long)b * N_NODES + i) * N_NODES;
        #pragma unroll
        for (int jj = 0; jj < 8; ++jj) Pr[jj * 32 + lane] = p[jj] * inv;
    }
}

// ---------------------------------------------------------------------------
// K3: out[b] = P[b][256,256] @ H[b][256,64]  (fp32 WMMA, K-loop of 64)
// H = columns 128..191 of S. One wave per 64x16 output block (4 i-tiles)
// so the strided H fragment is loaded once per k-step, reused by 4 WMMAs.
// ---------------------------------------------------------------------------
__global__ __launch_bounds__(32)
void gat_aggregate_wmma(const float* __restrict__ P,
                        const float* __restrict__ S,
                        float* __restrict__ out)
{
    const int lane = threadIdx.x & 31;
    const int half = lane >> 4;
    const int l16  = lane & 15;
    const int b     = blockIdx.y;
    const int iBlk  = blockIdx.x & 3;    // 0..3  (64 rows each)
    const int dTile = blockIdx.x >> 2;   // 0..3

    const float* Pb = P + (long)b * N_NODES * N_NODES;
    const float* Hb = S + (long)b * N_NODES * S_LD + 128;

    const int n = dTile * 16 + l16;
    const float* arow0 = Pb + (long)(iBlk * 64 + l16) * N_NODES;

    v8f c0 = {0.f,0.f,0.f,0.f,0.f,0.f,0.f,0.f};
    v8f c1 = c0, c2 = c0, c3 = c0;

    #pragma unroll 2
    for (int k0 = 0; k0 < N_NODES; k0 += 4) {
        const int kk = k0 + 2 * half;
        v2f bfrag;
        bfrag.x = Hb[(kk    ) * S_LD + n];
        bfrag.y = Hb[(kk + 1) * S_LD + n];

        v2f a0, a1, a2, a3;
        a0.x = arow0[kk];                 a0.y = arow0[kk + 1];
        a1.x = arow0[16 * N_NODES + kk];  a1.y = arow0[16 * N_NODES + kk + 1];
        a2.x = arow0[32 * N_NODES + kk];  a2.y = arow0[32 * N_NODES + kk + 1];
        a3.x = arow0[48 * N_NODES + kk];  a3.y = arow0[48 * N_NODES + kk + 1];

        c0 = __builtin_amdgcn_wmma_f32_16x16x4_f32(false, a0, false, bfrag, (short)0, c0, false, false);
        c1 = __builtin_amdgcn_wmma_f32_16x16x4_f32(false, a1, false, bfrag, (short)0, c1, false, false);
        c2 = __builtin_amdgcn_wmma_f32_16x16x4_f32(false, a2, false, bfrag, (short)0, c2, false, false);
        c3 = __builtin_amdgcn_wmma_f32_16x16x4_f32(false, a3, false, bfrag, (short)0, c3, false, false);
    }

    float* dst = out + ((long)b * N_NODES + iBlk * 64 + 8 * half) * N_OUT + dTile * 16 + l16;
    #pragma unroll
    for (int v = 0; v < 8; ++v) dst[(v     ) * N_OUT] = c0[v];
    #pragma unroll
    for (int v = 0; v < 8; ++v) dst[(v + 16) * N_OUT] = c1[v];
    #pragma unroll
    for (int v = 0; v < 8; ++v) dst[(v + 32) * N_OUT] = c2[v];
    #pragma unroll
    for (int v = 0; v < 8; ++v) dst[(v + 48) * N_OUT] = c3[v];
}

// ---------------------------------------------------------------------------
extern "C" void kernel_launch(void* const* d_in, const int* in_sizes, int n_in,
                              void* d_out, int out_size, void* d_ws, size_t ws_size,
                              hipStream_t stream) {
    (void)in_sizes; (void)n_in; (void)out_size; (void)ws_size;
    const float* x  = (const float*)d_in[0];
    // d_in[1] is `batch` (int32) — layout is implicit in the dense reshape, unused.
    const float* W  = (const float*)d_in[2];
    const float* Wl = (const float*)d_in[3];
    const float* a  = (const float*)d_in[4];
    float* out = (float*)d_out;

    float* S = (float*)d_ws;                                   // [8192,192] = 6 MB
    float* P = S + (size_t)(N_BATCH * N_NODES) * S_LD;         // [32,256,256] = 8 MB

    // K1: fused projections (s_i | s_j | h) via fp32 WMMA, 64x16 per wave
    gat_proj_wmma<<<dim3(128, 12), 32, 0, stream>>>(x, W, Wl, S);

    // K2: attention scores + softmax (VALU + LDS), one block per graph
    gat_att_softmax<<<dim3(N_BATCH), 256, N_OUT * N_NODES * sizeof(float), stream>>>(S, a, P);

    // K3: per-graph attention @ h via fp32 WMMA, 64x16 per wave
    gat_aggregate_wmma<<<dim3(16, N_BATCH), 32, 0, stream>>>(P, S, out);
}